// SprecherBlock_8727373546281
// MI455X (gfx1250) — compile-verified
//
#include <hip/hip_runtime.h>

#define D_DIM   256
#define Q_DIM   256
#define NKNOT   64
#define ALPHA_C 1.0f
#define PHI_XMIN_C  (-0.5f)
#define PHI_XMAX_C  ( 1.5f)
#define PHIB_XMIN_C (-10.0f)
#define PHIB_XMAX_C ( 10.0f)

typedef float v2f __attribute__((ext_vector_type(2)));
typedef float v8f __attribute__((ext_vector_type(8)));

// ---------------------------------------------------------------------------
// Prep: raw_increments -> normalized knot values ky, stored as overlapping
// pairs kp[i] = {ky[i], ky[i+1]} (63 float2) in workspace.
// ---------------------------------------------------------------------------
__global__ void sprecher_prep(const float* __restrict__ raw,
                              float* __restrict__ kp_out) {
    __shared__ float cs[NKNOT];
    int t = threadIdx.x;
    if (t < NKNOT) {
        float r  = raw[t];
        float sp = (r > 20.0f) ? r : log1pf(expf(r));   // softplus
        cs[t] = sp + 1e-6f;
    }
    __syncthreads();
    if (t == 0) {
        float run = 0.0f;
        for (int i = 0; i < NKNOT; ++i) { run += cs[i]; cs[i] = run; }  // cumsum
        float c0    = cs[0];
        float denom = (cs[NKNOT - 1] - c0) + 1e-12f;
        for (int i = 0; i < NKNOT; ++i) cs[i] = (cs[i] - c0) / denom;
    }
    __syncthreads();
    if (t < NKNOT - 1) {
        kp_out[2 * t]     = cs[t];
        kp_out[2 * t + 1] = cs[t + 1];
    }
}

// ---------------------------------------------------------------------------
// Inner monotone spline eval (returns clamped phi, NOT scaled by lam --
// lam rides in the WMMA B matrix).  8 VALU ops: add, floor, med3, sub,
// cvt, lshl, sub, fma(+clamp modifier) + one ds_load_b64 gather.
// clamp(y,0,1) == both reference `where` clamps since ky is strictly
// monotone 0 -> 1.
// ---------------------------------------------------------------------------
__device__ __forceinline__ float phi_eval(float u0, float qdel,
                                          const float2* kp) {
    float u  = u0 + qdel;
    float f  = floorf(u);
    float fc = __builtin_amdgcn_fmed3f(f, 0.0f, 62.0f);   // clip(floor(u),0,62)
    float t  = u - fc;
    int   i  = (int)fc;
    float2 k = kp[i];                                     // {ky[i], ky[i+1]}
    float y  = fmaf(t, k.y - k.x, k.x);
    y = __builtin_amdgcn_fmed3f(y, 0.0f, 1.0f);           // folds into fma clamp
    return y;
}

// ---------------------------------------------------------------------------
// Main: one block per batch row b.  8 waves; each wave owns 16 q's per chunk
// (2 chunks).  Reduction over d via V_WMMA_F32_16X16X4_F32:
//   A(16x4)  = phi values      (lanes 0-15: M=lane, K={0,1}; 16-31: K={2,3})
//   B(4x16)  = lam[d0..d0+3]   (VGPR0: K=0 / K=2 per half-wave; VGPR1: K=1/K=3)
//   D(16x16) accumulates s[b, q0+M] exactly in f32 (columns identical).
// ---------------------------------------------------------------------------
__global__ void __launch_bounds__(256)
sprecher_main(const float* __restrict__ x, const float* __restrict__ Phi_y,
              const float* __restrict__ lam, const float* __restrict__ eta_p,
              const float* __restrict__ kp_g, float* __restrict__ out) {
    __shared__ __align__(16) float2 u0lam[D_DIM];   // {u0[b,d], lam[d]}
    __shared__ float2 kp[NKNOT - 1];

    const float dxp   = (PHI_XMAX_C - PHI_XMIN_C) / (float)(NKNOT - 1);
    const float invdx = 1.0f / dxp;       // +1e-12 is below f32 ulp of dxp

    int   b   = blockIdx.x;
    int   tid = threadIdx.x;
    float eta = eta_p[0];

    {   // per-(b,d) precompute: u0 = (x - xmin)/dx, paired with lam[d]
        float xv = x[b * D_DIM + tid];
        float2 v;
        v.x = (xv - PHI_XMIN_C) * invdx;
        v.y = lam[tid];
        u0lam[tid] = v;
    }
    if (tid < NKNOT - 1) kp[tid] = ((const float2*)kp_g)[tid];
    __syncthreads();

    int lane  = tid & 31;
    int wave  = tid >> 5;
    int m     = lane & 15;       // q row within tile (M index)
    int hi    = lane >> 4;       // 0: K={0,1}   1: K={2,3}

    float qscale = eta * invdx;  // u = u0 + q * eta/dx

    const float4* u4 = (const float4*)u0lam;   // {u0[2p],lam[2p],u0[2p+1],lam[2p+1]}

    for (int chunk = wave; chunk < 16; chunk += 8) {
        int   q0   = chunk * 16;
        float qdel = (float)(q0 + m) * qscale;
        v8f acc = {};
        #pragma unroll 4
        for (int d0 = 0; d0 < D_DIM; d0 += 4) {
            // one ds_load_b128 broadcast: both {u0,lam} pairs for this half-wave
            float4 ul = u4[(d0 >> 1) + hi];
            v2f a, bm;
            a.x  = phi_eval(ul.x, qdel, kp);
            a.y  = phi_eval(ul.z, qdel, kp);
            bm.x = ul.y;                       // lam[d0+selHi]
            bm.y = ul.w;                       // lam[d0+selHi+1]
            // D = A x B + C  =>  acc[M][*] += sum_k phi_k * lam_k
            acc = __builtin_amdgcn_wmma_f32_16x16x4_f32(
                false, a, false, bm, (short)0, acc, false, false);
        }
        // D layout: lanes 0-15 hold M=j in VGPR j, lanes 16-31 hold M=8+j.
        // All columns identical; lanes 0 and 16 write the 16 outputs.
        if (m == 0) {
            const float dxb  = (PHIB_XMAX_C - PHIB_XMIN_C) / (float)(NKNOT - 1);
            const float invb = 1.0f / dxb;
            int base = q0 + hi * 8;
            #pragma unroll
            for (int j = 0; j < 8; ++j) {
                int   qq = base + j;
                float s  = acc[j] + ALPHA_C * (float)qq;
                // outer spline: interior lerp == both extrapolation branches
                float u  = (s - PHIB_XMIN_C) * invb;
                float f  = floorf(u);
                float fc = __builtin_amdgcn_fmed3f(f, 0.0f, 62.0f);
                float t  = u - fc;
                int   i  = (int)fc;
                float pL = Phi_y[i];
                float pR = Phi_y[i + 1];
                out[b * Q_DIM + qq] = fmaf(t, pR - pL, pL);
            }
        }
    }
}

// ---------------------------------------------------------------------------
extern "C" void kernel_launch(void* const* d_in, const int* in_sizes, int n_in,
                              void* d_out, int out_size, void* d_ws, size_t ws_size,
                              hipStream_t stream) {
    const float* x     = (const float*)d_in[0];
    const float* raw   = (const float*)d_in[1];
    const float* Phi_y = (const float*)d_in[2];
    const float* lam   = (const float*)d_in[3];
    const float* eta   = (const float*)d_in[4];
    float*       out   = (float*)d_out;
    float*       kp    = (float*)d_ws;       // 63 float2 = 504 bytes

    int B = in_sizes[0] / D_DIM;

    sprecher_prep<<<1, 64, 0, stream>>>(raw, kp);
    sprecher_main<<<B, 256, 0, stream>>>(x, Phi_y, lam, eta, kp, out);
}